// Distance_24524263260583
// MI455X (gfx1250) — compile-verified
//
#include <hip/hip_runtime.h>
#include <stdint.h>

// Periodic radius-graph, fixed K=32 nearest neighbors under minimum-image metric.
// Faithful to the JAX reference:
//   - selection metric: min-image d2, cutoff^2 = 36, self-loop included
//   - stable top-K (ties -> lower index), pad with self-loops
//   - emitted vec/weight use RAW (unwrapped) position diffs; self/pad -> 0
//
// CDNA5 strategy:
//   * Stage the full 96 KB position array into LDS per block with async
//     global->LDS DMA (ASYNCcnt-tracked global_load_async_to_lds_b128).
//   * 4-way split of the j-scan per atom (1024 wave32s total for occupancy);
//     segment is wave-uniform so LDS reads stay broadcast/conflict-free.
//   * Exact merge of the 4 sorted per-segment lists through LDS scratch that
//     overlays the position array (barrier-separated reuse).

#define N_ATOMS 8192
#define KNBR    32
#define SEGS    4
#define APB     64                    // atoms per block
#define SEG_LEN (N_ATOMS / SEGS)      // 2048
#define BLOCK_T (APB * SEGS)          // 256 threads = 8 wave32
#define CUT2    36.0f
#define BOX_L   50.0f
#define INV_BOX (1.0f / 50.0f)

__global__ __launch_bounds__(BLOCK_T) void
radius_graph_topk_kernel(const float* __restrict__ pos, float* __restrict__ out)
{
    // 96 KB LDS: positions during scan, candidate scratch (64 KB) during merge.
    __shared__ float smem[N_ATOMS * 3];

    const int tid = threadIdx.x;
    const int s   = tid >> 6;                 // segment 0..3, uniform per wave32
    const int a   = tid & (APB - 1);          // atom within block
    const int i   = blockIdx.x * APB + a;     // global atom id

    // ---- Stage all positions into LDS via CDNA5 async global->LDS DMA ----
    // 98304 B = 6144 x 16B chunks; 256 lanes -> 24 async B128 ops per lane.
    {
        const uint32_t lds_base = (uint32_t)(uintptr_t)(void*)smem;
        const int nChunks = (N_ATOMS * 3 * 4) / 16;   // 6144
        for (int c = tid; c < nChunks; c += BLOCK_T) {
            uint32_t goff  = (uint32_t)c * 16u;
            uint32_t laddr = lds_base + goff;
            asm volatile("global_load_async_to_lds_b128 %0, %1, %2"
                         :
                         : "v"(laddr), "v"(goff), "s"(pos)
                         : "memory");
        }
        asm volatile("s_wait_asynccnt 0x0" ::: "memory");
    }
    __syncthreads();

    const float xi = smem[3 * i + 0];
    const float yi = smem[3 * i + 1];
    const float zi = smem[3 * i + 2];

    // Per-segment sorted (ascending d2) top-K; init = +inf / self (pad value).
    float bd[KNBR];
    int   bi[KNBR];
#pragma unroll
    for (int k = 0; k < KNBR; ++k) { bd[k] = __builtin_inff(); bi[k] = i; }

    // Scan this thread's quarter of all atoms. j is wave-uniform -> LDS
    // broadcast reads. Strict '<' + ascending j gives stable tie ordering.
    const int j0 = s * SEG_LEN;
#pragma unroll 4
    for (int jj = 0; jj < SEG_LEN; ++jj) {
        const int j  = j0 + jj;
        const int j3 = 3 * j;
        float dx = xi - smem[j3 + 0];
        float dy = yi - smem[j3 + 1];
        float dz = zi - smem[j3 + 2];
        // minimum image; rintf = round-half-even = jnp.round
        float mx = dx - rintf(dx * INV_BOX) * BOX_L;
        float my = dy - rintf(dy * INV_BOX) * BOX_L;
        float mz = dz - rintf(dz * INV_BOX) * BOX_L;
        float d2 = mx * mx + my * my + mz * mz;
        if (d2 <= CUT2 && d2 < bd[KNBR - 1]) {
#pragma unroll
            for (int k = KNBR - 1; k >= 1; --k) {
                bool  sh  = d2 < bd[k - 1];            // shift incumbent up
                bool  pl  = (!sh) && (d2 < bd[k]);     // place new here
                float nbd = sh ? bd[k - 1] : (pl ? d2 : bd[k]);
                int   nbi = sh ? bi[k - 1] : (pl ? j  : bi[k]);
                bd[k] = nbd;
                bi[k] = nbi;
            }
            if (d2 < bd[0]) { bd[0] = d2; bi[0] = j; }
        }
    }

    // ---- Publish per-segment candidates into LDS scratch (overlays positions) ----
    __syncthreads();   // all scans done before positions are overwritten

    float* cd = smem;                                   // 8192 floats (32 KB)
    int*   ci = (int*)(smem + APB * SEGS * KNBR);       // 8192 ints   (32 KB)
    const int wbase = a * (SEGS * KNBR) + s * KNBR;
#pragma unroll
    for (int k = 0; k < KNBR; ++k) { cd[wbase + k] = bd[k]; ci[wbase + k] = bi[k]; }

    __syncthreads();

    // ---- Merge (one thread per atom; s==0 is uniform per wave -> no divergence) ----
    if (s == 0) {
        float md[KNBR];
        int   mi[KNBR];
#pragma unroll
        for (int k = 0; k < KNBR; ++k) { md[k] = __builtin_inff(); mi[k] = i; }

        const int cbase = a * (SEGS * KNBR);
        for (int c = 0; c < SEGS * KNBR; ++c) {         // segment-major order:
            float d   = cd[cbase + c];                  // preserves stable tie order
            int   idx = ci[cbase + c];
            if (d < md[KNBR - 1]) {                     // also skips (inf,i) pads
#pragma unroll
                for (int k = KNBR - 1; k >= 1; --k) {
                    bool  sh = d < md[k - 1];
                    bool  pl = (!sh) && (d < md[k]);
                    float nd = sh ? md[k - 1] : (pl ? d   : md[k]);
                    int   ni = sh ? mi[k - 1] : (pl ? idx : mi[k]);
                    md[k] = nd;
                    mi[k] = ni;
                }
                if (d < md[0]) { md[0] = d; mi[0] = idx; }
            }
        }

        // ---- Emit outputs (raw diffs from global pos; indices exact in fp32) ----
        const int NK = N_ATOMS * KNBR;
        float* srcRow = out;            // edge_index row 0
        float* dstRow = out + NK;       // edge_index row 1
        float* wRow   = out + 2 * NK;   // edge_weight
        float* vRow   = out + 3 * NK;   // edge_vec [NK,3]

#pragma unroll
        for (int k = 0; k < KNBR; ++k) {
            int   j  = mi[k];
            float vx = 0.f, vy = 0.f, vz = 0.f, w = 0.f;
            if (j != i) {
                vx = xi - pos[3 * j + 0];   // RAW (unwrapped) difference
                vy = yi - pos[3 * j + 1];
                vz = zi - pos[3 * j + 2];
                w  = sqrtf(vx * vx + vy * vy + vz * vz);
            }
            int r = i * KNBR + k;
            srcRow[r]       = (float)i;
            dstRow[r]       = (float)j;
            wRow[r]         = w;
            vRow[3 * r + 0] = vx;
            vRow[3 * r + 1] = vy;
            vRow[3 * r + 2] = vz;
        }
    }
}

extern "C" void kernel_launch(void* const* d_in, const int* in_sizes, int n_in,
                              void* d_out, int out_size, void* d_ws, size_t ws_size,
                              hipStream_t stream)
{
    (void)in_sizes; (void)n_in; (void)d_ws; (void)ws_size; (void)out_size;
    const float* pos = (const float*)d_in[0];   // [8192,3] float32
    float*       out = (float*)d_out;           // 6*N*K floats
    radius_graph_topk_kernel<<<N_ATOMS / APB, BLOCK_T, 0, stream>>>(pos, out);
}